// SJLTProjection_49632642072605
// MI455X (gfx1250) — compile-verified
//
#include <hip/hip_runtime.h>
#include <stdint.h>

// SJLT projection: y[b, idx[d,j]] += sign[d,j] * x[b,d]; y *= 1/sqrt(4)
#define BATCH    4096
#define ORIG_DIM 16384
#define PROJ_DIM 1024
#define ROWS     16                      // batch rows per workgroup
#define CHUNK    512                     // d-columns per TDM tile
#define NCHUNK   (ORIG_DIM / CHUNK)      // 32
#define XROW_STRIDE (CHUNK + CHUNK/256)  // 514: TDM pad inserts 1 dword / 256 dwords
#define YROW_STRIDE (PROJ_DIM + 1)       // 1025: bank-skew for conflict-free ds_add_f32

typedef uint32_t v4u __attribute__((ext_vector_type(4)));
typedef int      v8i __attribute__((ext_vector_type(8)));
typedef int      v4i __attribute__((ext_vector_type(4)));

// Group1 dword0 flag sets (ISA cdna5 §8.4)
#define TDM_F32               (2u << 16)                 // data_size = 4B
#define TDM_F32_PADDED        (TDM_F32 | (1u << 20) | (7u << 22))
// pad_enable, pad_interval=256 dwords, pad_amount field 0 => +1 dword / 256 dwords

// Generic 2D TENSOR_LOAD_TO_LDS (1D tiles: tile_d1 = 1).
__device__ __forceinline__ void tdm_load_2d(const void* gsrc, uint32_t lds_off,
                                            uint32_t tensor_d0, uint32_t tensor_d1,
                                            uint32_t tile_d0, uint32_t tile_d1,
                                            uint32_t stride_d0, uint32_t flags) {
  uint64_t ga = (uint64_t)(uintptr_t)gsrc;
  v4u g0;
  g0.x = 1u;                                            // count=1, user descriptor
  g0.y = lds_off;                                       // LDS byte offset
  g0.z = (uint32_t)ga;                                  // global_addr[31:0]
  g0.w = ((uint32_t)(ga >> 32) & 0x01FFFFFFu)           // global_addr[56:32]
       | (2u << 30);                                    // type=2 ("image")
  v8i g1;
  g1[0] = (int)flags;
  g1[1] = (int)((tensor_d0 & 0xFFFFu) << 16);                          // tensor_dim0[15:0]
  g1[2] = (int)(((tensor_d0 >> 16) & 0xFFFFu) | ((tensor_d1 & 0xFFFFu) << 16));
  g1[3] = (int)(((tensor_d1 >> 16) & 0xFFFFu) | (tile_d0 << 16));      // tile_dim0
  g1[4] = (int)(tile_d1 & 0xFFFFu);                                    // tile_dim1, tile_dim2=0
  g1[5] = (int)stride_d0;                                              // tensor_dim0_stride[31:0]
  g1[6] = 0;
  g1[7] = 0;
  v4i g2 = (v4i)(0);                                    // dims 2/3 unused
  v4i g3 = (v4i)(0);
  v8i g4 = (v8i)(0);                                    // extra group (clang-23 6-arg form)
  __builtin_amdgcn_tensor_load_to_lds(g0, g1, g2, g3, g4, 0);
}

__global__ __launch_bounds__(256)
void sjlt_scatter_kernel(const float* __restrict__ x,
                         const int*   __restrict__ idx,   // [ORIG_DIM][4]
                         const int*   __restrict__ sgn,   // [ORIG_DIM][4], values +-1
                         float*       __restrict__ out) { // [BATCH][PROJ_DIM]
  __shared__ float yacc[ROWS * YROW_STRIDE];                  // 65.6 KB accumulator
  __shared__ float xbuf[2][ROWS * XROW_STRIDE];               // 2 x 32.9 KB x tiles
  __shared__ __align__(16) int ibuf[2][CHUNK * 4];            // 2 x 8 KB index chunks
  __shared__ __align__(16) int sbuf[2][CHUNK * 4];            // 2 x 8 KB sign chunks

  const int t       = threadIdx.x;
  const int wave    = t >> 5;
  const int rowBase = blockIdx.x * ROWS;
  const float* xtile = x + (size_t)rowBase * ORIG_DIM;

  // zero the accumulator (first in-loop barrier publishes it)
  for (int i = t; i < ROWS * YROW_STRIDE; i += 256) yacc[i] = 0.0f;

  // prologue: wave 0 kicks off chunk 0 (x tile + idx + sgn) via the TDM
  if (wave == 0) {
    tdm_load_2d(xtile, (uint32_t)(uintptr_t)&xbuf[0][0],
                ORIG_DIM, BATCH, CHUNK, ROWS, ORIG_DIM, TDM_F32_PADDED);
    tdm_load_2d(idx, (uint32_t)(uintptr_t)&ibuf[0][0],
                ORIG_DIM * 4, 1, CHUNK * 4, 1, ORIG_DIM * 4, TDM_F32);
    tdm_load_2d(sgn, (uint32_t)(uintptr_t)&sbuf[0][0],
                ORIG_DIM * 4, 1, CHUNK * 4, 1, ORIG_DIM * 4, TDM_F32);
  }

  const int r = t & 15;   // batch row within tile (bank-skewed atomics across lanes)
  const int g = t >> 4;   // column group

  for (int k = 0; k < NCHUNK; ++k) {
    if (wave == 0) {
      if (k + 1 < NCHUNK) {
        const int p = (k + 1) & 1;
        const int dNext = (k + 1) * CHUNK;
        tdm_load_2d(xtile + dNext, (uint32_t)(uintptr_t)&xbuf[p][0],
                    ORIG_DIM, BATCH, CHUNK, ROWS, ORIG_DIM, TDM_F32_PADDED);
        tdm_load_2d(idx + dNext * 4, (uint32_t)(uintptr_t)&ibuf[p][0],
                    ORIG_DIM * 4, 1, CHUNK * 4, 1, ORIG_DIM * 4, TDM_F32);
        tdm_load_2d(sgn + dNext * 4, (uint32_t)(uintptr_t)&sbuf[p][0],
                    ORIG_DIM * 4, 1, CHUNK * 4, 1, ORIG_DIM * 4, TDM_F32);
        __builtin_amdgcn_s_wait_tensorcnt(3);  // chunk k's 3 transfers landed; k+1 in flight
      } else {
        __builtin_amdgcn_s_wait_tensorcnt(0);
      }
    }
    __syncthreads();  // publish TDM-filled buffers to all waves

    const float* xrow = &xbuf[k & 1][r * XROW_STRIDE];
    const int*   ib   = &ibuf[k & 1][0];
    const int*   sb   = &sbuf[k & 1][0];
    #pragma unroll 4
    for (int c = g; c < CHUNK; c += 16) {
      // padded tile layout: element (r,c) at r*514 + c + c/256 -> conflict-free reads
      const float xv = xrow[c + (c >> 8)];
      const int4  ix = *reinterpret_cast<const int4*>(ib + c * 4);  // LDS broadcast
      const int4  sg = *reinterpret_cast<const int4*>(sb + c * 4);  // LDS broadcast
      float* yr = &yacc[r * YROW_STRIDE];   // stride 1025 => lanes hit distinct banks
      atomicAdd(&yr[ix.x], xv * (float)sg.x);   // ds_add_f32 (no return)
      atomicAdd(&yr[ix.y], xv * (float)sg.y);
      atomicAdd(&yr[ix.z], xv * (float)sg.z);
      atomicAdd(&yr[ix.w], xv * (float)sg.w);
    }
    __syncthreads();  // buffers (k&1) free for reuse before next issue
  }

  // scaled write-out, fully coalesced
  const float norm = 0.5f;  // 1/sqrt(C), C=4
  for (int f = t; f < ROWS * PROJ_DIM; f += 256) {
    const int row = f >> 10;
    const int col = f & (PROJ_DIM - 1);
    out[(size_t)(rowBase + row) * PROJ_DIM + col] = yacc[row * YROW_STRIDE + col] * norm;
  }
}

extern "C" void kernel_launch(void* const* d_in, const int* in_sizes, int n_in,
                              void* d_out, int out_size, void* d_ws, size_t ws_size,
                              hipStream_t stream) {
  const float* x   = (const float*)d_in[0];
  const int*   idx = (const int*)d_in[1];
  const int*   sgn = (const int*)d_in[2];
  float*       out = (float*)d_out;
  (void)in_sizes; (void)n_in; (void)out_size; (void)d_ws; (void)ws_size;

  dim3 grid(BATCH / ROWS);   // 256 workgroups
  dim3 block(256);           // 8 wave32s
  hipLaunchKernelGGL(sjlt_scatter_kernel, grid, block, 0, stream, x, idx, sgn, out);
}